// FGPool1_88656714924911
// MI455X (gfx1250) — compile-verified
//
#include <hip/hip_runtime.h>
#include <hip/hip_bf16.h>
#include <math.h>

// ---------------- problem constants ----------------
#define Bg   200
#define NPER 500
#define NN   100000          // Bg*NPER
#define EE   3200000         // NN*32
#define DD   128
#define KK   400             // kept per graph
#define BK   (Bg*KK)         // 80000

static constexpr size_t ND = (size_t)NN * DD;    // 12,800,000

// ---------------- workspace layout (float elements) ----------------
static constexpr size_t WS_A    = 0;              // agg  [N,D]
static constexpr size_t WS_XT   = ND;             // xt   [N,D]
static constexpr size_t WS_XTP  = 2 * ND;         // x_tp [N,D]
static constexpr size_t WS_XTN  = 3 * ND;         // x_tn [N,D]
static constexpr size_t WS_TP   = 4 * ND;                 // [N]
static constexpr size_t WS_TN   = WS_TP + NN;
static constexpr size_t WS_UP   = WS_TN + NN;
static constexpr size_t WS_UN   = WS_UP + NN;
static constexpr size_t WS_AGGP = WS_UN + NN;
static constexpr size_t WS_AGGN = WS_AGGP + NN;
static constexpr size_t WS_SPP  = WS_AGGN + NN;
static constexpr size_t WS_SNP  = WS_SPP + NN;
static constexpr size_t WS_RO   = WS_SNP + NN;    // readouts [3,B,D]
static constexpr size_t WS_TANH = WS_RO + 3 * (size_t)Bg * DD;  // [B*K]
static constexpr size_t WS_PERM = WS_TANH + BK;   // int region [B*K]
static constexpr size_t WS_MAP  = WS_PERM + BK;   // int region [N]

// ---------------- output layout (float elements, concatenated in return order)
static constexpr size_t OUT_X     = 0;                         // [B*K, D]
static constexpr size_t OUT_EDGE  = (size_t)BK * DD;           // [2, E]
static constexpr size_t OUT_BATCH = OUT_EDGE + 2 * (size_t)EE; // [B*K]
static constexpr size_t OUT_PERM  = OUT_BATCH + BK;            // [B*K]
static constexpr size_t OUT_LOSS  = OUT_PERM + BK;             // scalar

typedef float v2f __attribute__((ext_vector_type(2)));
typedef float v8f __attribute__((ext_vector_type(8)));

__device__ __forceinline__ float lrelu(float v) { return v > 0.f ? v : 0.2f * v; }
__device__ __forceinline__ float softplus_stable(float x) {
    return fmaxf(x, 0.f) + log1pf(expf(-fabsf(x)));
}

// ---------------- kernel 0: zero accumulators (agg matrix, scalar aggs, loss)
__global__ void k_zero(float* __restrict__ ws, float* __restrict__ out) {
    size_t stride = (size_t)gridDim.x * blockDim.x;
    size_t total  = ND + 2 * (size_t)NN;
    for (size_t t = (size_t)blockIdx.x * blockDim.x + threadIdx.x; t < total; t += stride) {
        if (t < ND) ws[WS_A + t] = 0.f;
        else        ws[WS_AGGP + (t - ND)] = 0.f;   // AGGP then AGGN (contiguous)
    }
    if (blockIdx.x == 0 && threadIdx.x == 0) out[OUT_LOSS] = 0.f;
}

// ---------------- kernel 1: agg[dst] += x[src]  (one wave lane = 4 dims of one edge)
__global__ void k_agg(const float* __restrict__ x, const int* __restrict__ src,
                      const int* __restrict__ dst, float* __restrict__ A) {
    size_t tid = (size_t)blockIdx.x * blockDim.x + threadIdx.x;
    size_t e = tid >> 5;
    int lane = (int)(tid & 31);
    if (e >= (size_t)EE) return;
    int s = src[e], d = dst[e];
    const float4 v = *(const float4*)(x + (size_t)s * DD + lane * 4);
    float* o = A + (size_t)d * DD + lane * 4;
    atomicAdd(o + 0, v.x); atomicAdd(o + 1, v.y);
    atomicAdd(o + 2, v.z); atomicAdd(o + 3, v.w);
}

// ---------------- kernel 2: fused triple GraphConv via f32 WMMA
// out_c = lrelu(A @ Wr_c + X @ Wl_c + br_c), c in {transform, pp, np}
// One wave -> one 16x16 output tile. K-loop: 32+32 V_WMMA_F32_16X16X4_F32.
__global__ void k_conv3(const float* __restrict__ A, const float* __restrict__ X,
                        const float* __restrict__ Wr0, const float* __restrict__ Wl0,
                        const float* __restrict__ br0, float* __restrict__ out0,
                        const float* __restrict__ Wr1, const float* __restrict__ Wl1,
                        const float* __restrict__ br1, float* __restrict__ out1,
                        const float* __restrict__ Wr2, const float* __restrict__ Wl2,
                        const float* __restrict__ br2, float* __restrict__ out2) {
    size_t wave = ((size_t)blockIdx.x * blockDim.x + threadIdx.x) >> 5;
    int lane = threadIdx.x & 31;
    const int ROWTILES = NN / 16;                  // 6250, exact
    int row_tile = (int)(wave / 24);
    int r        = (int)(wave % 24);
    int conv = r >> 3;                             // 0..2
    int col  = r & 7;                              // 0..7 (128/16 col tiles)
    if (row_tile >= ROWTILES) return;              // wave-uniform

    const float* Wr = conv == 0 ? Wr0 : (conv == 1 ? Wr1 : Wr2);
    const float* Wl = conv == 0 ? Wl0 : (conv == 1 ? Wl1 : Wl2);
    const float* br = conv == 0 ? br0 : (conv == 1 ? br1 : br2);
    float*      out = conv == 0 ? out0 : (conv == 1 ? out1 : out2);

    int mlo  = lane & 15;            // M (A rows) / N (B cols) index per lane
    int khalf = (lane >> 4) * 2;     // lanes 0-15 hold K=k0,k0+1; 16-31 hold K=k0+2,k0+3
    size_t m = (size_t)row_tile * 16 + mlo;
    int n = col * 16 + mlo;

    v8f acc = {};
    // agg @ Wr
    #pragma unroll 4
    for (int k0 = 0; k0 < DD; k0 += 4) {
        const float* ap = A + m * DD + k0 + khalf;
        v2f a; a.x = ap[0]; a.y = ap[1];
        const float* bp = Wr + (size_t)(k0 + khalf) * DD + n;
        v2f b; b.x = bp[0]; b.y = bp[DD];
        acc = __builtin_amdgcn_wmma_f32_16x16x4_f32(false, a, false, b,
                                                    (short)0, acc, false, false);
    }
    // x @ Wl
    #pragma unroll 4
    for (int k0 = 0; k0 < DD; k0 += 4) {
        const float* ap = X + m * DD + k0 + khalf;
        v2f a; a.x = ap[0]; a.y = ap[1];
        const float* bp = Wl + (size_t)(k0 + khalf) * DD + n;
        v2f b; b.x = bp[0]; b.y = bp[DD];
        acc = __builtin_amdgcn_wmma_f32_16x16x4_f32(false, a, false, b,
                                                    (short)0, acc, false, false);
    }
    // C/D layout: VGPR i: lanes0-15 -> M=i, lanes16-31 -> M=i+8; N = lane&15 (+ col*16)
    float bias = br[n];
    int mbase = row_tile * 16 + 8 * (lane >> 4);
    #pragma unroll
    for (int i = 0; i < 8; ++i) {
        size_t row = (size_t)(mbase + i);
        out[row * DD + n] = lrelu(acc[i] + bias);
    }
}

// ---------------- kernel 3: scalar projections for the dim-1 pooling convs
// tp = x_tp@WrP, up = x_tp@WlP, tn = x_tn@WrN, un = x_tn@WlN
__global__ void k_proj(const float* __restrict__ xtp, const float* __restrict__ xtn,
                       const float* __restrict__ WrP, const float* __restrict__ WlP,
                       const float* __restrict__ WrN, const float* __restrict__ WlN,
                       float* __restrict__ tp, float* __restrict__ up,
                       float* __restrict__ tn, float* __restrict__ un) {
    int i = blockIdx.x * blockDim.x + threadIdx.x;
    if (i >= NN) return;
    const float* p = xtp + (size_t)i * DD;
    const float* q = xtn + (size_t)i * DD;
    float a = 0.f, b = 0.f, c = 0.f, d = 0.f;
    #pragma unroll 8
    for (int k = 0; k < DD; ++k) {
        float pv = p[k], qv = q[k];
        a += pv * WrP[k]; b += pv * WlP[k];
        c += qv * WrN[k]; d += qv * WlN[k];
    }
    tp[i] = a; up[i] = b; tn[i] = c; un[i] = d;
}

// ---------------- kernel 4: scalar segment sums (projection commutes with segsum)
__global__ void k_sagg(const int* __restrict__ src, const int* __restrict__ dst,
                       const float* __restrict__ tp, const float* __restrict__ tn,
                       float* __restrict__ aggp, float* __restrict__ aggn) {
    int e = blockIdx.x * blockDim.x + threadIdx.x;
    if (e >= EE) return;
    int s = src[e], d = dst[e];
    atomicAdd(&aggp[d], tp[s]);
    atomicAdd(&aggn[d], tn[s]);
}

// ---------------- kernel 5: finalize node scores
__global__ void k_score(const float* __restrict__ aggp, const float* __restrict__ aggn,
                        const float* __restrict__ up, const float* __restrict__ un,
                        const float* __restrict__ brP, const float* __restrict__ brN,
                        float* __restrict__ spp, float* __restrict__ snp) {
    int i = blockIdx.x * blockDim.x + threadIdx.x;
    if (i >= NN) return;
    spp[i] = aggp[i] + brP[0] + up[i];
    snp[i] = aggn[i] + brN[0] + un[i];
}

// ---------------- kernel 6: per-graph gated mean readouts (block = graph, thread = dim)
__global__ void k_readout(const float* __restrict__ xt, const float* __restrict__ spp,
                          const float* __restrict__ snp, float* __restrict__ ro) {
    __shared__ float stp[NPER], stn[NPER];
    int g = blockIdx.x, d = threadIdx.x;
    for (int i = d; i < NPER; i += DD) {
        int node = g * NPER + i;
        stp[i] = tanhf(spp[node]);
        stn[i] = tanhf(snp[node]);
    }
    __syncthreads();
    float sp = 0.f, sn = 0.f, sr = 0.f;
    for (int i = 0; i < NPER; ++i) {
        float v = xt[((size_t)g * NPER + i) * DD + d];
        sr += v; sp += v * stp[i]; sn += v * stn[i];
    }
    const float inv = 1.f / NPER;
    ro[((size_t)0 * Bg + g) * DD + d] = sp * inv;   // x_pp_readout
    ro[((size_t)1 * Bg + g) * DD + d] = sn * inv;   // x_np_readout
    ro[((size_t)2 * Bg + g) * DD + d] = sr * inv;   // x_readout
}

// ---------------- kernel 7: discriminator + BCE loss (block = graph, thread = hidden unit)
__global__ void k_disc(const float* __restrict__ ro, const float* __restrict__ W1,
                       const float* __restrict__ b1, const float* __restrict__ W2,
                       const float* __restrict__ b2, float* __restrict__ loss) {
    __shared__ float sp[DD], sn[DD];
    int g = blockIdx.x, j = threadIdx.x;
    const float* rpp = ro + ((size_t)0 * Bg + g) * DD;
    const float* rnp = ro + ((size_t)1 * Bg + g) * DD;
    const float* rr  = ro + ((size_t)2 * Bg + g) * DD;
    float hp = 0.f, hn = 0.f;
    #pragma unroll 4
    for (int i = 0; i < DD; ++i) {
        float w = W1[(size_t)i * DD + j];
        hp += rpp[i] * w; hn += rnp[i] * w;
    }
    #pragma unroll 4
    for (int i = 0; i < DD; ++i) {
        float w = W1[(size_t)(DD + i) * DD + j];
        float v = rr[i];
        hp += v * w; hn += v * w;
    }
    hp = lrelu(hp + b1[j]); hn = lrelu(hn + b1[j]);
    sp[j] = hp * W2[j]; sn[j] = hn * W2[j];
    __syncthreads();
    for (int s = DD / 2; s > 0; s >>= 1) {
        if (j < s) { sp[j] += sp[j + s]; sn[j] += sn[j + s]; }
        __syncthreads();
    }
    if (j == 0) {
        float zp = sp[0] + b2[0], zn = sn[0] + b2[0];
        float contrib = 0.5f * (softplus_stable(-zp) + softplus_stable(zn)) / (float)Bg;
        atomicAdd(loss, contrib);
    }
}

// ---------------- kernel 8: exact top-K per graph (rank by value desc, index asc)
__global__ void k_topk(const float* __restrict__ spp, const float* __restrict__ snp,
                       float* __restrict__ out, float* __restrict__ tanhvals,
                       int* __restrict__ permi, int* __restrict__ mapping) {
    __shared__ float sc[NPER];
    int g = blockIdx.x, i = threadIdx.x;
    if (i < NPER) sc[i] = spp[g * NPER + i] - snp[g * NPER + i];
    __syncthreads();
    if (i >= NPER) return;
    float v = sc[i];
    int rank = 0;
    for (int j = 0; j < NPER; ++j) {
        float u = sc[j];
        rank += (u > v) || (u == v && j < i);
    }
    int node = g * NPER + i;
    if (rank < KK) {
        int row = g * KK + rank;
        mapping[node] = row;
        permi[row]    = node;
        tanhvals[row] = tanhf(v);
        out[OUT_PERM + row]  = (float)node;   // exact: node < 2^24
        out[OUT_BATCH + row] = (float)g;
    } else {
        mapping[node] = -1;
    }
}

// ---------------- kernel 9: x_out[row] = xt[perm[row]] * tanh(top_val)
__global__ void k_gather(const float* __restrict__ xt, const int* __restrict__ permi,
                         const float* __restrict__ tanhvals, float* __restrict__ out) {
    size_t tid = (size_t)blockIdx.x * blockDim.x + threadIdx.x;
    size_t row = tid >> 5;
    int lane = (int)(tid & 31);
    if (row >= (size_t)BK) return;
    int node = permi[row];
    float s = tanhvals[row];
    const float4 v = *(const float4*)(xt + (size_t)node * DD + lane * 4);
    float4 o = { v.x * s, v.y * s, v.z * s, v.w * s };
    *(float4*)(out + OUT_X + row * DD + lane * 4) = o;
}

// ---------------- kernel 10: filter_adj relabel
__global__ void k_edgefilt(const int* __restrict__ src, const int* __restrict__ dst,
                           const int* __restrict__ mapping, float* __restrict__ out) {
    int e = blockIdx.x * blockDim.x + threadIdx.x;
    if (e >= EE) return;
    int ns = mapping[src[e]], nd = mapping[dst[e]];
    bool valid = (ns >= 0) && (nd >= 0);
    out[OUT_EDGE + e]            = valid ? (float)ns : -1.f;
    out[OUT_EDGE + (size_t)EE + e] = valid ? (float)nd : -1.f;
}

extern "C" void kernel_launch(void* const* d_in, const int* in_sizes, int n_in,
                              void* d_out, int out_size, void* d_ws, size_t ws_size,
                              hipStream_t stream) {
    // Inputs in setup_inputs() insertion order (params leaves in nested dict order).
    const float* x    = (const float*)d_in[0];
    const int*   src  = (const int*)d_in[1];
    const int*   dst  = src + EE;
    // d_in[2] = batch (unused: graphs are equal-sized, layout is implicit)
    const float* tWr  = (const float*)d_in[3];
    const float* tbr  = (const float*)d_in[4];
    const float* tWl  = (const float*)d_in[5];
    const float* pWr  = (const float*)d_in[6];
    const float* pbr  = (const float*)d_in[7];
    const float* pWl  = (const float*)d_in[8];
    const float* nWr  = (const float*)d_in[9];
    const float* nbr  = (const float*)d_in[10];
    const float* nWl  = (const float*)d_in[11];
    const float* posWr = (const float*)d_in[12];
    const float* posbr = (const float*)d_in[13];
    const float* posWl = (const float*)d_in[14];
    const float* negWr = (const float*)d_in[15];
    const float* negbr = (const float*)d_in[16];
    const float* negWl = (const float*)d_in[17];
    const float* dW1  = (const float*)d_in[18];
    const float* db1  = (const float*)d_in[19];
    const float* dW2  = (const float*)d_in[20];
    const float* db2  = (const float*)d_in[21];

    float* ws  = (float*)d_ws;
    float* out = (float*)d_out;
    float* A   = ws + WS_A;
    float* xt  = ws + WS_XT;
    float* xtp = ws + WS_XTP;
    float* xtn = ws + WS_XTN;
    int*   permi   = (int*)(ws + WS_PERM);
    int*   mapping = (int*)(ws + WS_MAP);

    // 0) zero accumulators + loss
    k_zero<<<2048, 256, 0, stream>>>(ws, out);

    // 1) shared D-dim segment sum (one pass serves all three convs)
    {
        size_t threads = (size_t)EE * 32;
        k_agg<<<(unsigned)((threads + 255) / 256), 256, 0, stream>>>(x, src, dst, A);
    }

    // 2) fused 3x GraphConv GEMM on the WMMA pipes
    {
        size_t waves = (size_t)(NN / 16) * 24;     // 150,000 waves, exact
        k_conv3<<<(unsigned)(waves * 32 / 256), 256, 0, stream>>>(
            A, x,
            tWr, tWl, tbr, xt,
            pWr, pWl, pbr, xtp,
            nWr, nWl, nbr, xtn);
    }

    // 3) scalar projections for the two dim-1 pooling convs
    k_proj<<<(NN + 255) / 256, 256, 0, stream>>>(
        xtp, xtn, posWr, posWl, negWr, negWl,
        ws + WS_TP, ws + WS_UP, ws + WS_TN, ws + WS_UN);

    // 4) scalar segment sums (128x less atomic traffic than vector aggregation)
    k_sagg<<<(EE + 255) / 256, 256, 0, stream>>>(
        src, dst, ws + WS_TP, ws + WS_TN, ws + WS_AGGP, ws + WS_AGGN);

    // 5) s_pp / s_np
    k_score<<<(NN + 255) / 256, 256, 0, stream>>>(
        ws + WS_AGGP, ws + WS_AGGN, ws + WS_UP, ws + WS_UN,
        posbr, negbr, ws + WS_SPP, ws + WS_SNP);

    // 6) per-graph readouts
    k_readout<<<Bg, DD, 0, stream>>>(xt, ws + WS_SPP, ws + WS_SNP, ws + WS_RO);

    // 7) discriminator + loss
    k_disc<<<Bg, DD, 0, stream>>>(ws + WS_RO, dW1, db1, dW2, db2, out + OUT_LOSS);

    // 8) exact top-K selection per graph
    k_topk<<<Bg, 512, 0, stream>>>(ws + WS_SPP, ws + WS_SNP, out,
                                   ws + WS_TANH, permi, mapping);

    // 9) gated gather of kept node features
    {
        size_t threads = (size_t)BK * 32;
        k_gather<<<(unsigned)((threads + 255) / 256), 256, 0, stream>>>(
            xt, permi, ws + WS_TANH, out);
    }

    // 10) edge filtering / relabeling
    k_edgefilt<<<(EE + 255) / 256, 256, 0, stream>>>(src, dst, mapping, out);
}